// GraphConvolution_43224550868074
// MI455X (gfx1250) — compile-verified
//
#include <hip/hip_runtime.h>

// GCN layer on MI455X (gfx1250, wave32):
//   h = x @ W            -> V_WMMA_F32_16X16X4_F32 tiles (fp32-exact matrix path)
//   agg = scatter(edges) -> register segment-accumulate (rows sorted) + f32 atomics
//   out = relu(agg)      -> vectorized epilogue
//
// h (100000x256 f32 = 102 MB) fits in the 192 MB L2, so the 3.2 GB of edge
// gathers stream from L2 rather than HBM. Sorted edge_row lets each wave keep
// a running 256-wide accumulator in VGPRs and flush with global_atomic_add_f32
// only on row change (~16-32x atomic reduction vs per-edge atomics).

#define DIM 256          // D_IN == D_OUT == 256
#define EPW 32           // edges processed per wave in the SpMM

typedef float v2f __attribute__((ext_vector_type(2)));
typedef float v8f __attribute__((ext_vector_type(8)));

// ---------------------------------------------------------------------------
// Zero d_out (it is poisoned to 0xAA before timing).
// ---------------------------------------------------------------------------
__global__ __launch_bounds__(256) void zero_f4(float4* __restrict__ p, int n4) {
  int i = blockIdx.x * blockDim.x + threadIdx.x;
  if (i < n4) p[i] = make_float4(0.f, 0.f, 0.f, 0.f);
}

// ---------------------------------------------------------------------------
// Dense GEMM: H[M, 256] = X[M, 256] * W[256, 256], fp32, WMMA 16x16x4.
// Block = 256 threads = 8 waves; each wave owns a 16(M) x 64(N) strip:
// 4 accumulator tiles sharing one A-fragment per K-step (A reuse x4).
// grid = (ceil(M/128), 256/64).
//
// Lane layout per CDNA5 ISA:
//   A (16x4):  lanes 0-15 rows M=lane, VGPR0/1 = K{0,1}; lanes 16-31 = K{2,3}
//   B (4x16):  mirrored - lanes hold col N=(lane&15), VGPR0/1 = rows K{0,1}
//              (lanes 16-31: K{2,3})
//   C/D 16x16: VGPR v -> M = v (+8 for lanes 16-31), N = lane&15
// ---------------------------------------------------------------------------
__global__ __launch_bounds__(256) void gemm_f32_wmma(
    const float* __restrict__ X, const float* __restrict__ W,
    float* __restrict__ H, int M) {
  const int lane = threadIdx.x & 31;
  const int wave = threadIdx.x >> 5;
  const int hi   = lane >> 4;     // 0: lanes 0-15, 1: lanes 16-31
  const int l16  = lane & 15;

  const int m0 = blockIdx.x * 128 + wave * 16;
  const int n0 = blockIdx.y * 64;

  // A-fragment source row (clamped; out-of-range rows are masked at store).
  const int arow = m0 + l16;
  const float* aptr = X + (size_t)(arow < M ? arow : 0) * DIM;

  v8f acc0 = {}; v8f acc1 = {}; v8f acc2 = {}; v8f acc3 = {};

  const float* wbase = W + n0 + l16;

  #pragma unroll 4
  for (int k = 0; k < DIM; k += 4) {
    const int ka = k + 2 * hi;                 // this lane-half's K pair
    v2f a = *(const v2f*)(aptr + ka);          // A[arow][ka], A[arow][ka+1]

    const float* b0p = wbase + (size_t)ka * DIM;   // row ka
    const float* b1p = b0p + DIM;                  // row ka+1
    v2f b0; b0.x = b0p[0];  b0.y = b1p[0];
    v2f b1; b1.x = b0p[16]; b1.y = b1p[16];
    v2f b2; b2.x = b0p[32]; b2.y = b1p[32];
    v2f b3; b3.x = b0p[48]; b3.y = b1p[48];

    acc0 = __builtin_amdgcn_wmma_f32_16x16x4_f32(false, a, false, b0,
                                                 (short)0, acc0, false, false);
    acc1 = __builtin_amdgcn_wmma_f32_16x16x4_f32(false, a, false, b1,
                                                 (short)0, acc1, false, false);
    acc2 = __builtin_amdgcn_wmma_f32_16x16x4_f32(false, a, false, b2,
                                                 (short)0, acc2, false, false);
    acc3 = __builtin_amdgcn_wmma_f32_16x16x4_f32(false, a, false, b3,
                                                 (short)0, acc3, false, false);
  }

  // Store D tiles: VGPR v -> row m0 + hi*8 + v, col n0 + {0,16,32,48} + l16.
  const int mb = m0 + hi * 8;
  float* hrow = H + n0 + l16;
  #pragma unroll
  for (int v = 0; v < 8; ++v) {
    const int m = mb + v;
    if (m < M) {
      float* p = hrow + (size_t)m * DIM;
      p[0]  = acc0[v];
      p[16] = acc1[v];
      p[32] = acc2[v];
      p[48] = acc3[v];
    }
  }
}

// ---------------------------------------------------------------------------
// SpMM: out[row] += val * h[col], edge_row sorted ascending.
// One wave handles EPW consecutive edges; each lane owns features
// [8*lane, 8*lane+8). Accumulate in registers while the row is unchanged
// (row test is wave-uniform), flush with global_atomic_add_f32 on change.
// Gathers are two coalesced float4 loads -> one contiguous 1 KB line/edge,
// served from L2 since h fits entirely in the 192 MB L2.
// ---------------------------------------------------------------------------
__global__ __launch_bounds__(256) void spmm_seg(
    const float* __restrict__ H,
    const int* __restrict__ erow, const int* __restrict__ ecol,
    const float* __restrict__ eval,
    float* __restrict__ out, int nE) {
  const int lane = threadIdx.x & 31;
  const int wgid = blockIdx.x * (blockDim.x >> 5) + (threadIdx.x >> 5);
  const int e0 = wgid * EPW;
  if (e0 >= nE) return;
  const int e1 = (e0 + EPW < nE) ? (e0 + EPW) : nE;

  const int fbase = lane * 8;   // 8 contiguous features per lane

  float acc[8];
  #pragma unroll
  for (int j = 0; j < 8; ++j) acc[j] = 0.f;

  int cur = erow[e0];           // wave-uniform

  for (int e = e0; e < e1; ++e) {
    const int r = erow[e];      // wave-uniform
    if (r != cur) {             // uniform branch: flush finished segment
      float* o = out + (size_t)cur * DIM + fbase;
      #pragma unroll
      for (int j = 0; j < 8; ++j) atomicAdd(o + j, acc[j]);
      #pragma unroll
      for (int j = 0; j < 8; ++j) acc[j] = 0.f;
      cur = r;
    }
    const int   c = ecol[e];
    const float v = eval[e];
    const float4* hp = (const float4*)(H + (size_t)c * DIM + fbase);
    const float4 h0 = hp[0];
    const float4 h1 = hp[1];
    acc[0] += v * h0.x; acc[1] += v * h0.y;
    acc[2] += v * h0.z; acc[3] += v * h0.w;
    acc[4] += v * h1.x; acc[5] += v * h1.y;
    acc[6] += v * h1.z; acc[7] += v * h1.w;
  }

  // Flush the trailing segment (its row may continue in the next wave's
  // chunk -> atomics make cross-wave partial sums correct).
  float* o = out + (size_t)cur * DIM + fbase;
  #pragma unroll
  for (int j = 0; j < 8; ++j) atomicAdd(o + j, acc[j]);
}

// ---------------------------------------------------------------------------
// ReLU epilogue, vectorized.
// ---------------------------------------------------------------------------
__global__ __launch_bounds__(256) void relu_f4(float4* __restrict__ p, int n4) {
  int i = blockIdx.x * blockDim.x + threadIdx.x;
  if (i < n4) {
    float4 v = p[i];
    v.x = fmaxf(v.x, 0.f);
    v.y = fmaxf(v.y, 0.f);
    v.z = fmaxf(v.z, 0.f);
    v.w = fmaxf(v.w, 0.f);
    p[i] = v;
  }
}

// ---------------------------------------------------------------------------
// Launch: zero(out) -> gemm(h=x@W into d_ws) -> spmm(scatter into out) -> relu
// All on `stream`; d_ws holds h (M*256 f32 = ~102 MB).
// ---------------------------------------------------------------------------
extern "C" void kernel_launch(void* const* d_in, const int* in_sizes, int n_in,
                              void* d_out, int out_size, void* d_ws, size_t ws_size,
                              hipStream_t stream) {
  const float* x    = (const float*)d_in[0];
  const float* w    = (const float*)d_in[1];
  const int*   erow = (const int*)d_in[2];
  const int*   ecol = (const int*)d_in[3];
  const float* evl  = (const float*)d_in[4];
  float* out = (float*)d_out;
  float* h   = (float*)d_ws;

  const int M  = in_sizes[0] / DIM;   // number of nodes
  const int nE = in_sizes[2];         // number of edges

  // 1) zero the (poisoned) output accumulator
  const int n4 = out_size / 4;
  zero_f4<<<(n4 + 255) / 256, 256, 0, stream>>>((float4*)out, n4);

  // 2) dense GEMM via fp32 WMMA
  dim3 gg((M + 127) / 128, DIM / 64);
  gemm_f32_wmma<<<gg, 256, 0, stream>>>(x, w, h, M);

  // 3) segment-accumulating SpMM scatter
  const int waves  = (nE + EPW - 1) / EPW;
  const int blocks = (waves + 7) / 8;
  spmm_seg<<<blocks, 256, 0, stream>>>(h, erow, ecol, evl, out, nE);

  // 4) ReLU
  relu_f4<<<(n4 + 255) / 256, 256, 0, stream>>>((float4*)out, n4);
}